// GCNStack_73323681677413
// MI455X (gfx1250) — compile-verified
//
#include <hip/hip_runtime.h>

// GCN stack for MI455X (gfx1250, wave32). fp32 end-to-end, matmuls via
// V_WMMA_F32_16X16X4_F32. All GEMMs are HBM-bound (AI ~= 21 flop/byte for the
// edge update, less for node GEMMs), so fp32 WMMA is the right matrix op.
// Weights are staged in LDS *pre-paired* (v2f {W[2p][col],W[2p+1][col]}) so a
// B fragment is a single aligned ds_load_b64 straight into an even VGPR pair,
// eliminating the v_dual_mov shuffles seen with row-major LDS staging.

#define NN 100000
#define EE 1600000
#define H  64
#define NLAYERS 3
#define NHEADS 4
#define HD 16
#define LN_EPS 1e-5f

typedef __attribute__((ext_vector_type(2))) float v2f;
typedef __attribute__((ext_vector_type(8))) float v8f;

__device__ __forceinline__ v8f wmma4(v2f a, v2f b, v8f c) {
  // D = A(16x4 f32) * B(4x16 f32) + C(16x16 f32)
  return __builtin_amdgcn_wmma_f32_16x16x4_f32(
      /*neg_a=*/false, a, /*neg_b=*/false, b,
      /*c_mod=*/(short)0, c, /*reuse_a=*/false, /*reuse_b=*/false);
}

// Stage a K x 64 row-major weight matrix into LDS as K/2 x 64 v2f pairs:
// dst[p*64 + col] = { W[2p][col], W[2p+1][col] }.
__device__ __forceinline__ void stage_w_pairs(const float* __restrict__ W,
                                              v2f* __restrict__ dst, int kdim,
                                              int tid) {
  int npair = kdim >> 1;
  for (int i = tid; i < npair * 16; i += 256) {
    int p = i >> 4, j = (i & 15) * 4;
    float4 r0 = *(const float4*)(W + (size_t)(2 * p) * H + j);
    float4 r1 = *(const float4*)(W + (size_t)(2 * p + 1) * H + j);
    v2f t;
    t.x = r0.x; t.y = r1.x; dst[p * H + j + 0] = t;
    t.x = r0.y; t.y = r1.y; dst[p * H + j + 1] = t;
    t.x = r0.z; t.y = r1.z; dst[p * H + j + 2] = t;
    t.x = r0.w; t.y = r1.w; dst[p * H + j + 3] = t;
  }
}

// ---------------- degree / norm ----------------
__global__ void k_init_deg(float* __restrict__ deg) {
  int i = blockIdx.x * 256 + threadIdx.x;
  if (i < NN) deg[i] = 1.0f;
}
__global__ void k_deg_acc(const int* __restrict__ ei, float* __restrict__ deg) {
  int e = blockIdx.x * 256 + threadIdx.x;
  if (e < EE) atomicAdd(&deg[ei[EE + e]], 1.0f);
}
__global__ void k_rsqrt(const float* __restrict__ deg, float* __restrict__ dis) {
  int i = blockIdx.x * 256 + threadIdx.x;
  if (i < NN) dis[i] = rsqrtf(deg[i]);
}
__global__ void k_zero128(float* __restrict__ p) {
  p[threadIdx.x] = 0.0f;  // launched with 128 threads: sum_xp[64] + pooled[64]
}

// ---------------- xW = x @ gcn_w ; xnew = xW/deg + b (WMMA) ----------------
__global__ __launch_bounds__(256) void k_gcn_xw(
    const float* __restrict__ xin, const float* __restrict__ W,
    const float* __restrict__ bias, const float* __restrict__ deg,
    float* __restrict__ xw, float* __restrict__ xnew) {
  __shared__ v2f w_lds[32 * H];  // 16 KB, paired layout
  int tid = threadIdx.x;
  stage_w_pairs(W, w_lds, H, tid);
  __syncthreads();
  int wave = tid >> 5, lane = tid & 31;
  int row0 = blockIdx.x * 128 + wave * 16;
  if (row0 >= NN) return;                 // wave-uniform; NN % 16 == 0
  int hi = (lane >> 4) & 1, m = lane & 15;
  int row = row0 + m;
  const float* xr = xin + (size_t)row * H;
  v2f a[16];
#pragma unroll
  for (int c = 0; c < 16; ++c) {          // A fragments: lane half picks K pair
    int k = 4 * c + 2 * hi;
    v2f t; t.x = xr[k]; t.y = xr[k + 1]; a[c] = t;
  }
#pragma unroll
  for (int t4 = 0; t4 < 4; ++t4) {
    v8f acc = {};
    int col = t4 * 16 + m;
#pragma unroll
    for (int c = 0; c < 16; ++c) {
      v2f b = w_lds[(2 * c + hi) * H + col];  // single ds_load_b64
      acc = wmma4(a[c], b, acc);
    }
#pragma unroll
    for (int v = 0; v < 8; ++v) {
      int orow = row0 + v + 8 * hi;       // C/D layout: M = v (+8 for hi half)
      float val = acc[v];
      xw[(size_t)orow * H + col] = val;
      xnew[(size_t)orow * H + col] = val / deg[orow] + bias[col];
    }
  }
}

// ---------------- message scatter: xnew[dst] += norm * xw[src] ----------------
__global__ __launch_bounds__(256) void k_scatter(
    const int* __restrict__ ei, const float* __restrict__ xw,
    const float* __restrict__ dis, float* __restrict__ xnew) {
  int e = blockIdx.x * 256 + threadIdx.x;
  if (e >= EE) return;
  int s = ei[e], d = ei[EE + e];
  float nrm = dis[s] * dis[d];
  const float4* xr = (const float4*)(xw + (size_t)s * H);
  float* out = xnew + (size_t)d * H;
#pragma unroll
  for (int j = 0; j < 16; ++j) {
    float4 v = xr[j];
    atomicAdd(out + 4 * j + 0, v.x * nrm);
    atomicAdd(out + 4 * j + 1, v.y * nrm);
    atomicAdd(out + 4 * j + 2, v.z * nrm);
    atomicAdd(out + 4 * j + 3, v.w * nrm);
  }
}

// -------- xp = xnew @ proj_w + proj_b (WMMA) + column sums for head-mean --------
__global__ __launch_bounds__(256) void k_proj(
    const float* __restrict__ xin, const float* __restrict__ W,
    const float* __restrict__ bias, float* __restrict__ xp,
    float* __restrict__ sum_xp) {
  __shared__ v2f w_lds[32 * H];
  __shared__ float ssum[H];
  int tid = threadIdx.x;
  stage_w_pairs(W, w_lds, H, tid);
  if (tid < H) ssum[tid] = 0.0f;
  __syncthreads();
  int wave = tid >> 5, lane = tid & 31;
  int row0 = blockIdx.x * 128 + wave * 16;
  int hi = (lane >> 4) & 1, m = lane & 15;
  if (row0 < NN) {                        // wave-uniform guard, barriers outside
    int row = row0 + m;
    const float* xr = xin + (size_t)row * H;
    v2f a[16];
#pragma unroll
    for (int c = 0; c < 16; ++c) {
      int k = 4 * c + 2 * hi;
      v2f t; t.x = xr[k]; t.y = xr[k + 1]; a[c] = t;
    }
#pragma unroll
    for (int t4 = 0; t4 < 4; ++t4) {
      v8f acc = {};
      int col = t4 * 16 + m;
#pragma unroll
      for (int c = 0; c < 16; ++c) {
        v2f b = w_lds[(2 * c + hi) * H + col];
        acc = wmma4(a[c], b, acc);
      }
      float csum = 0.0f;
#pragma unroll
      for (int v = 0; v < 8; ++v) {
        int orow = row0 + v + 8 * hi;
        float val = acc[v] + bias[col];
        xp[(size_t)orow * H + col] = val;
        csum += val;
      }
      atomicAdd(&ssum[col], csum);
    }
  }
  __syncthreads();
  if (tid < H) atomicAdd(&sum_xp[tid], ssum[tid]);
}

// -------- per-node head scores, softmax over 4 heads, weighted pooling --------
__global__ __launch_bounds__(256) void k_pool(
    const float* __restrict__ xp, const float* __restrict__ xnew,
    const float* __restrict__ sum_xp, float* __restrict__ pooled) {
  __shared__ float smean[H];
  __shared__ float sp[H];
  int tid = threadIdx.x;
  if (tid < H) { smean[tid] = sum_xp[tid] * (1.0f / NN); sp[tid] = 0.0f; }
  __syncthreads();
  int n = blockIdx.x * 256 + tid;
  if (n < NN) {
    const float* xr = xp + (size_t)n * H;
    float w[NHEADS];
    float mx = -3.0e38f;
#pragma unroll
    for (int h = 0; h < NHEADS; ++h) {
      float s = 0.0f;
#pragma unroll
      for (int d2 = 0; d2 < HD; ++d2) s += xr[h * HD + d2] * smean[h * HD + d2];
      w[h] = s; mx = fmaxf(mx, s);
    }
    float sum = 0.0f;
#pragma unroll
    for (int h = 0; h < NHEADS; ++h) { w[h] = __expf(w[h] - mx); sum += w[h]; }
    float inv = 1.0f / sum;
    const float* nr = xnew + (size_t)n * H;
#pragma unroll
    for (int j = 0; j < H; ++j) atomicAdd(&sp[j], nr[j] * w[j >> 4] * inv);
  }
  __syncthreads();
  if (tid < H) atomicAdd(&pooled[tid], sp[tid]);
}

// -------- layernorm(64) + relu on the pooled vector (single 64-thread block) --------
__global__ void k_ln(const float* __restrict__ pooled, const float* __restrict__ g,
                     const float* __restrict__ b, float* __restrict__ pf) {
  __shared__ float red[64];
  int t = threadIdx.x;
  float v = pooled[t];
  red[t] = v; __syncthreads();
  for (int s = 32; s > 0; s >>= 1) { if (t < s) red[t] += red[t + s]; __syncthreads(); }
  float mu = red[0] * (1.0f / 64.0f);
  __syncthreads();
  float dv = v - mu;
  red[t] = dv * dv; __syncthreads();
  for (int s = 32; s > 0; s >>= 1) { if (t < s) red[t] += red[t + s]; __syncthreads(); }
  float var = red[0] * (1.0f / 64.0f);
  float out = dv * rsqrtf(var + LN_EPS) * g[t] + b[t];
  pf[t] = fmaxf(out, 0.0f);
}

// -------- x = pooled + x (broadcast residual) --------
__global__ void k_resid(const float* __restrict__ xin, const float* __restrict__ pf,
                        float* __restrict__ xout) {
  int i = blockIdx.x * 256 + threadIdx.x;
  if (i < NN * H) xout[i] = xin[i] + pf[i & 63];
}

// -------- edge update: out_e = [emb[src] | emb[dst]] @ eu_w + eu_b (WMMA, K=128) --------
__global__ __launch_bounds__(256) void k_edge(
    const float* __restrict__ ein, const int* __restrict__ ei,
    const float* __restrict__ W, const float* __restrict__ bias,
    float* __restrict__ eout) {
  __shared__ v2f w_lds[64 * H];  // 128x64 paired = 32 KB (WGP has 320 KB)
  int tid = threadIdx.x;
  stage_w_pairs(W, w_lds, 2 * H, tid);
  __syncthreads();
  int wave = tid >> 5, lane = tid & 31;
  int hi = (lane >> 4) & 1, m = lane & 15;
  int row0 = blockIdx.x * 128 + wave * 16;   // EE % 128 == 0: always full tiles
  int row = row0 + m;
  int s = ei[row], d = ei[EE + row];
  const float* srow = ein + (size_t)s * H;
  const float* drow = ein + (size_t)d * H;
  v2f a[32];
#pragma unroll
  for (int c = 0; c < 32; ++c) {
    int k = 4 * c + 2 * hi;                  // pairs stay within src/dst half
    const float* base = (k < H) ? (srow + k) : (drow + (k - H));
    v2f t; t.x = base[0]; t.y = base[1]; a[c] = t;  // -> global_load_b64 gather
  }
#pragma unroll
  for (int t4 = 0; t4 < 4; ++t4) {
    v8f acc = {};
    int col = t4 * 16 + m;
#pragma unroll
    for (int c = 0; c < 32; ++c) {
      v2f b = w_lds[(2 * c + hi) * H + col];  // single ds_load_b64
      acc = wmma4(a[c], b, acc);
    }
#pragma unroll
    for (int v = 0; v < 8; ++v) {
      int orow = row0 + v + 8 * hi;
      eout[(size_t)orow * H + col] = acc[v] + bias[col];
    }
  }
}

extern "C" void kernel_launch(void* const* d_in, const int* in_sizes, int n_in,
                              void* d_out, int out_size, void* d_ws, size_t ws_size,
                              hipStream_t stream) {
  (void)in_sizes; (void)n_in; (void)out_size; (void)ws_size;
  const float* x0     = (const float*)d_in[0];
  const int*   ei     = (const int*)d_in[1];
  const float* e0     = (const float*)d_in[2];
  const float* gcn_w  = (const float*)d_in[3];
  const float* gcn_b  = (const float*)d_in[4];
  const float* ln_g   = (const float*)d_in[5];
  const float* ln_b   = (const float*)d_in[6];
  const float* proj_w = (const float*)d_in[7];
  const float* proj_b = (const float*)d_in[8];
  const float* eu_w   = (const float*)d_in[9];
  const float* eu_b   = (const float*)d_in[10];

  float* ws     = (float*)d_ws;
  float* deg    = ws;
  float* dis    = ws + NN;
  float* sum_xp = ws + 2 * NN;        // 64
  float* pooled = sum_xp + 64;        // 64
  float* pf     = pooled + 64;        // 64
  float* xbuf   = ws + 2 * NN + 256;
  float* xw     = xbuf + (size_t)NN * H;
  float* xnew   = xw   + (size_t)NN * H;
  float* xp     = xnew + (size_t)NN * H;
  float* ebuf   = xp   + (size_t)NN * H;   // E*H floats

  float* out_x = (float*)d_out;
  float* out_e = out_x + (size_t)NN * H;

  k_init_deg<<<(NN + 255) / 256, 256, 0, stream>>>(deg);
  k_deg_acc<<<(EE + 255) / 256, 256, 0, stream>>>(ei, deg);
  k_rsqrt<<<(NN + 255) / 256, 256, 0, stream>>>(deg, dis);

  int gemm_blocks = (NN + 127) / 128;
  int edge_blocks = EE / 128;
  for (int i = 0; i < NLAYERS; ++i) {
    const float* xin  = (i == 0) ? x0 : xbuf;
    float*       xout = (i == NLAYERS - 1) ? out_x : xbuf;
    // edge ping-pong: in -> out_e -> ebuf -> out_e
    const float* ein  = (i == 0) ? e0 : ((i == 1) ? out_e : ebuf);
    float*       eout = (i == 1) ? ebuf : out_e;

    k_gcn_xw<<<gemm_blocks, 256, 0, stream>>>(xin, gcn_w + i * H * H,
                                              gcn_b + i * H, deg, xw, xnew);
    k_scatter<<<(EE + 255) / 256, 256, 0, stream>>>(ei, xw, dis, xnew);
    k_zero128<<<1, 128, 0, stream>>>(sum_xp);
    k_proj<<<gemm_blocks, 256, 0, stream>>>(xnew, proj_w + i * H * H,
                                            proj_b + i * H, xp, sum_xp);
    k_pool<<<(NN + 255) / 256, 256, 0, stream>>>(xp, xnew, sum_xp, pooled);
    k_ln<<<1, 64, 0, stream>>>(pooled, ln_g + i * H, ln_b + i * H, pf);
    k_resid<<<(NN * H + 255) / 256, 256, 0, stream>>>(xin, pf, xout);
    k_edge<<<edge_blocks, 256, 0, stream>>>(ein, ei, eu_w + (size_t)i * 2 * H * H,
                                            eu_b + i * H, eout);
  }
}